// SWTLayer_38740605010207
// MI455X (gfx1250) — compile-verified
//
#include <hip/hip_runtime.h>

// ---------------------------------------------------------------------------
// db2 SWT level-1 along channels as a banded GEMM on the f32 WMMA pipe.
//   out[b, o,     h, w] = sum_j LO[j] * x[b, (o+2-j) & 63, h, w]   (cA, o<64)
//   out[b, 64+o,  h, w] = sum_j HI[j] * x[b, (o+2-j) & 63, h, w]   (cD)
// W[o][c] = f[(o+2-c) & 63] if that index < 4 else 0  -> circulant, banded.
// GEMM tiling: M=16 (out ch), K=4 (in ch), N=16 (contiguous w) with
// V_WMMA_F32_16X16X4_F32. Only 6 K-tiles per M-tile are nonzero, and the
// A-fragments depend only on (o-c) mod 64, so they are M-tile invariant:
// 12 fragments total, precomputed once per wave.
// ---------------------------------------------------------------------------

typedef __attribute__((ext_vector_type(2))) float v2f;
typedef __attribute__((ext_vector_type(8))) float v8f;

#define BATCH 8
#define CIN   64
#define HDIM  128
#define WDIM  128
#define HW    (HDIM * WDIM)       // 16384

__device__ __forceinline__ float tap4(int j, float f0, float f1, float f2, float f3) {
  // select f[j] for j in [0,4), else 0  (4 cndmasks, no memory access)
  float v = 0.0f;
  v = (j == 0) ? f0 : v;
  v = (j == 1) ? f1 : v;
  v = (j == 2) ? f2 : v;
  v = (j == 3) ? f3 : v;
  return v;
}

__global__ __launch_bounds__(256) void swt_db2_wmma(const float* __restrict__ x,
                                                    float* __restrict__ out) {
  // pywt db2 dec filters, norm=True (divided by sqrt(2)), in float32
  const float SQ2 = 1.41421356237309504880f;
  const float LO0 = -0.12940952255092145f / SQ2;
  const float LO1 =  0.22414386804185735f / SQ2;
  const float LO2 =  0.836516303737469f   / SQ2;
  const float LO3 =  0.48296291314469025f / SQ2;
  const float HI0 = -0.48296291314469025f / SQ2;
  const float HI1 =  0.836516303737469f   / SQ2;
  const float HI2 = -0.22414386804185735f / SQ2;
  const float HI3 = -0.12940952255092145f / SQ2;

  const int tid  = threadIdx.x;
  const int wave = tid >> 5;          // 8 waves per block
  const int lane = tid & 31;
  const int r    = lane & 15;         // row-in-tile / w-in-strip
  const int ka   = (lane >> 4) << 1;  // K sub-offset per lane half: 0 or 2

  // one 16-wide w-strip per wave: strips = 8 * 128 * (128/16) = 8192
  const int strip = blockIdx.x * 8 + wave;
  const int w0 = (strip & 7) << 4;
  const int hb = strip >> 3;
  const int h  = hb & (HDIM - 1);
  const int b  = hb >> 7;

  // ---- load this lane's 32 needed input channels (each element read once) --
  const float* xb = x + (size_t)b * CIN * HW + (size_t)h * WDIM + w0 + r;
  float xv[32];
#pragma unroll
  for (int s = 0; s < 16; ++s) {
    const int c0 = 4 * s + ka;
    xv[2 * s]     = xb[(size_t)c0 * HW];
    xv[2 * s + 1] = xb[(size_t)(c0 + 1) * HW];
  }

  // ---- precompute the 12 M-tile-invariant A fragments ---------------------
  // For M-tile t, K-tile s = (4t + ds) & 15, ds in [-1..4]:
  //   j = (r + 2 - 4*ds - (ka+q)) & 63, weight = f[j] if j<4 else 0.
  float aLoX[6], aLoY[6], aHiX[6], aHiY[6];
#pragma unroll
  for (int i = 0; i < 6; ++i) {
    const int ds = i - 1;
    const int j0 = (r + 2 - 4 * ds - ka) & 63;
    const int j1 = (r + 2 - 4 * ds - (ka + 1)) & 63;
    aLoX[i] = tap4(j0, LO0, LO1, LO2, LO3);
    aLoY[i] = tap4(j1, LO0, LO1, LO2, LO3);
    aHiX[i] = tap4(j0, HI0, HI1, HI2, HI3);
    aHiY[i] = tap4(j1, HI0, HI1, HI2, HI3);
  }

  // ---- 4 M-tiles x (lo,hi) halves, 6 WMMAs each ---------------------------
#pragma unroll
  for (int t = 0; t < 4; ++t) {
    v8f accLo = {};
    v8f accHi = {};
#pragma unroll
    for (int i = 0; i < 6; ++i) {
      const int s = (4 * t + i - 1) & 15;
      v2f bf;  bf.x  = xv[2 * s];  bf.y  = xv[2 * s + 1];
      v2f alo; alo.x = aLoX[i];    alo.y = aLoY[i];
      v2f ahi; ahi.x = aHiX[i];    ahi.y = aHiY[i];
      // D = A x B + C ; 8-arg form (neg_a, A, neg_b, B, c_mod, C, reuse_a, reuse_b)
      accLo = __builtin_amdgcn_wmma_f32_16x16x4_f32(
          false, alo, false, bf, (short)0, accLo, false, false);
      accHi = __builtin_amdgcn_wmma_f32_16x16x4_f32(
          false, ahi, false, bf, (short)0, accHi, false, false);
    }
    // C/D layout: VGPR q holds M = q (lanes 0-15) / M = q+8 (lanes 16-31)
    const int mTop = 16 * t + ((lane >> 4) << 3);
    const size_t baseLo =
        ((size_t)b * 2 * CIN + mTop) * HW + (size_t)h * WDIM + w0 + r;
    const size_t baseHi = baseLo + (size_t)CIN * HW;
#pragma unroll
    for (int q = 0; q < 8; ++q) {
      out[baseLo + (size_t)q * HW] = accLo[q];
      out[baseHi + (size_t)q * HW] = accHi[q];
    }
  }
}

extern "C" void kernel_launch(void* const* d_in, const int* in_sizes, int n_in,
                              void* d_out, int out_size, void* d_ws, size_t ws_size,
                              hipStream_t stream) {
  (void)in_sizes; (void)n_in; (void)d_ws; (void)ws_size; (void)out_size;
  const float* x = (const float*)d_in[0];
  float* out = (float*)d_out;
  // 8192 strips / 8 waves per block = 1024 blocks of 256 threads (8 wave32)
  swt_db2_wmma<<<dim3(1024), dim3(256), 0, stream>>>(x, out);
}